// NormDistBase_944892805267
// MI455X (gfx1250) — compile-verified
//
#include <hip/hip_runtime.h>
#include <stdint.h>

// norm_dist_inf: out[b,o,n] = max_d |x[b,d,n] - w[o,d]| + bias[o]
// B=64, CIN=1024, COUT=1024, N=49
// VALU-bound (max-of-abs-diff contraction; WMMA inapplicable).
// CDNA5 path: TDM tensor_load_to_lds stages 16x1024 f32 weight tile into LDS.

#define B_    64
#define CIN_  1024
#define COUT_ 1024
#define NSP_  49
#define OT_   16   // output channels per block (weight tile rows)
#define BT_   5    // batches per block
#define NTH_  256

typedef uint32_t u32;
typedef uint64_t u64;
typedef __attribute__((ext_vector_type(4))) u32 v4u;
typedef __attribute__((ext_vector_type(8))) u32 v8u;

__global__ __launch_bounds__(NTH_)
void norm_dist_inf_kernel(const float* __restrict__ x,
                          const float* __restrict__ w,
                          const float* __restrict__ bias,
                          float* __restrict__ out) {
    __shared__ float wtile[OT_ * CIN_];   // 64 KB LDS weight tile

    const int o_tile = blockIdx.x * OT_;
    const int b_base = blockIdx.y * BT_;

    // ---- TDM: one DMA of w[o_tile .. o_tile+15][0..1023] into LDS ----
    if (threadIdx.x == 0) {
        u64 ga = (u64)(uintptr_t)(w + (size_t)o_tile * CIN_);
        // Flat LDS address truncates to the LDS byte offset in addr[31:0].
        u32 lds_base = (u32)(uintptr_t)&wtile[0];

        v4u g0;
        g0[0] = 1u;                                   // count=1 (valid), user D#
        g0[1] = lds_base;                             // lds_addr
        g0[2] = (u32)ga;                              // global_addr[31:0]
        g0[3] = (u32)((ga >> 32) & 0x1FFFFFFu)        // global_addr[56:32]
              | (2u << 30);                           // type = 2 ("image")

        v8u g1;
        g1[0] = (2u << 16);                           // data_size = 2 -> 4 bytes
        g1[1] = ((u32)(CIN_ & 0xFFFF)) << 16;         // tensor_dim0 lo16 (1024)
        g1[2] = ((u32)(CIN_ >> 16))                   // tensor_dim0 hi16 (0)
              | (((u32)(OT_ & 0xFFFF)) << 16);        // tensor_dim1 lo16 (16)
        g1[3] = ((u32)(CIN_ & 0xFFFF)) << 16;         // tensor_dim1 hi16=0 | tile_dim0=1024
        g1[4] = (u32)OT_;                             // tile_dim1 = 16 rows, tile_dim2 = 0
        g1[5] = (u32)CIN_;                            // tensor_dim0_stride lo32 = 1024
        g1[6] = 0u;                                   // stride hi | dim1_stride lo
        g1[7] = 0u;                                   // dim1_stride hi

        asm volatile("tensor_load_to_lds %0, %1" :: "s"(g0), "s"(g1) : "memory");
        __builtin_amdgcn_s_wait_tensorcnt(0);
    }
    __syncthreads();

    // ---- thread -> (b_local, n) mapping: 245/256 lanes active ----
    const int t = threadIdx.x;
    const int b_local = t / NSP_;
    const int n = t - b_local * NSP_;
    const int b = b_base + b_local;
    if (b_local >= BT_ || b >= B_) return;

    const float* __restrict__ xp = x + (size_t)b * CIN_ * NSP_ + n;

    float acc[OT_];
#pragma unroll
    for (int o = 0; o < OT_; ++o) acc[o] = 0.0f;   // result >= 0 always

    for (int d = 0; d < CIN_; d += 4) {
        // 4 coalesced x loads (consecutive n across lanes), reused by 16 o's
        const float xv0 = xp[(size_t)(d + 0) * NSP_];
        const float xv1 = xp[(size_t)(d + 1) * NSP_];
        const float xv2 = xp[(size_t)(d + 2) * NSP_];
        const float xv3 = xp[(size_t)(d + 3) * NSP_];
#pragma unroll
        for (int o = 0; o < OT_; ++o) {
            // ds_load_b128, same address in all lanes -> LDS broadcast
            const float4 wv = *(const float4*)&wtile[o * CIN_ + d];
            // v_sub_f32 + v_max_f32 with |src| modifier: 2 VALU per element
            acc[o] = fmaxf(acc[o], fabsf(xv0 - wv.x));
            acc[o] = fmaxf(acc[o], fabsf(xv1 - wv.y));
            acc[o] = fmaxf(acc[o], fabsf(xv2 - wv.z));
            acc[o] = fmaxf(acc[o], fabsf(xv3 - wv.w));
        }
    }

    float* __restrict__ op = out + ((size_t)b * COUT_ + o_tile) * NSP_ + n;
#pragma unroll
    for (int o = 0; o < OT_; ++o) {
        op[(size_t)o * NSP_] = acc[o] + bias[o_tile + o];
    }
}

extern "C" void kernel_launch(void* const* d_in, const int* in_sizes, int n_in,
                              void* d_out, int out_size, void* d_ws, size_t ws_size,
                              hipStream_t stream) {
    const float* x    = (const float*)d_in[0];
    const float* wgt  = (const float*)d_in[1];
    const float* bias = (const float*)d_in[2];
    float* out = (float*)d_out;

    dim3 grid(COUT_ / OT_, (B_ + BT_ - 1) / BT_);   // (64, 13)
    norm_dist_inf_kernel<<<grid, NTH_, 0, stream>>>(x, wgt, bias, out);
}